// DirectionalMultiHeadClassifier_15272903704882
// MI455X (gfx1250) — compile-verified
//
#include <hip/hip_runtime.h>
#include <hip/hip_bf16.h>

// ---------------------------------------------------------------------------
// DirectionalMultiHeadClassifier for MI455X (gfx1250, wave32)
//   B=16, S=2048, H=1024
//   Phase 1: bandwidth-bound masked reductions over hidden (128 MiB, ~5.8us floor)
//   Phase 2: tiny GEMM heads via V_WMMA_F32_16X16X4_F32 (exact f32 math)
// ---------------------------------------------------------------------------

#define B_  16
#define S_  2048
#define H_  1024
#define NSEG 16
#define SEGLEN (S_ / NSEG)   // 128

typedef float v2f __attribute__((ext_vector_type(2)));
typedef float v8f __attribute__((ext_vector_type(8)));

__device__ __forceinline__ float gelu_exact(float x) {
    return 0.5f * x * (1.0f + erff(x * 0.70710678118654752440f));
}

// ---------------------------------------------------------------------------
// K0: per-batch sequence length  L[b] = sum_s mask[b][s]
// ---------------------------------------------------------------------------
__global__ __launch_bounds__(256) void lengths_kernel(const int* __restrict__ mask,
                                                      int* __restrict__ Lout) {
    __shared__ int red[256];
    const int b = blockIdx.x, tid = threadIdx.x;
    int s = 0;
    for (int i = tid; i < S_; i += 256) s += mask[b * S_ + i];
    red[tid] = s;
    __syncthreads();
    for (int off = 128; off > 0; off >>= 1) {
        if (tid < off) red[tid] += red[tid + off];
        __syncthreads();
    }
    if (tid == 0) Lout[b] = red[0];
}

// ---------------------------------------------------------------------------
// K1: streaming masked partial sums.  grid = (NSEG, B), 256 threads.
// Each thread owns a float4 slice of H; weights per s cached in LDS.
// part layout: [seg][b][4][H]  (float)
// ---------------------------------------------------------------------------
__global__ __launch_bounds__(256) void partial_sums_kernel(
    const float* __restrict__ hidden, const int* __restrict__ mask,
    const int* __restrict__ Lin, float* __restrict__ part) {
    __shared__ float4 wts[SEGLEN];
    const int seg = blockIdx.x;   // 0..15
    const int b   = blockIdx.y;   // 0..15
    const int tid = threadIdx.x;  // 0..255
    const int s0  = seg * SEGLEN;

    const int L   = Lin[b];
    const int mid = L >> 1;
    const int Lm1 = L - 1;
    int st = L - 64; if (st < 1) st = 1;

    if (tid < SEGLEN) {
        const int s = s0 + tid;
        float4 w;
        w.x = (float)mask[b * S_ + s];                    // pooled weight
        w.y = (s >= 1   && s < mid) ? 1.0f : 0.0f;        // first half
        w.z = (s >= mid && s < Lm1) ? 1.0f : 0.0f;        // second half
        w.w = (s >= st  && s < Lm1) ? 1.0f : 0.0f;        // ending window
        wts[tid] = w;
    }
    __syncthreads();

    const float4* hp = (const float4*)hidden + (size_t)(b * S_ + s0) * (H_ / 4) + tid;
    float4 a0 = {0, 0, 0, 0}, a1 = a0, a2 = a0, a3 = a0;
    for (int s = 0; s < SEGLEN; ++s) {
        const float4 x = hp[(size_t)s * (H_ / 4)];
        const float4 w = wts[s];
        a0.x = fmaf(x.x, w.x, a0.x); a0.y = fmaf(x.y, w.x, a0.y);
        a0.z = fmaf(x.z, w.x, a0.z); a0.w = fmaf(x.w, w.x, a0.w);
        a1.x = fmaf(x.x, w.y, a1.x); a1.y = fmaf(x.y, w.y, a1.y);
        a1.z = fmaf(x.z, w.y, a1.z); a1.w = fmaf(x.w, w.y, a1.w);
        a2.x = fmaf(x.x, w.z, a2.x); a2.y = fmaf(x.y, w.z, a2.y);
        a2.z = fmaf(x.z, w.z, a2.z); a2.w = fmaf(x.w, w.z, a2.w);
        a3.x = fmaf(x.x, w.w, a3.x); a3.y = fmaf(x.y, w.w, a3.y);
        a3.z = fmaf(x.z, w.w, a3.z); a3.w = fmaf(x.w, w.w, a3.w);
    }
    float4* p4 = (float4*)part + (size_t)((seg * B_ + b) * 4) * (H_ / 4) + tid;
    p4[0 * (H_ / 4)] = a0;
    p4[1 * (H_ / 4)] = a1;
    p4[2 * (H_ / 4)] = a2;
    p4[3 * (H_ / 4)] = a3;
}

// ---------------------------------------------------------------------------
// K2: reduce segments (fixed order), means, layernorm, build head inputs.
// X layout: [4][b][H]:  X0=relu(second-first) X1=relu(first-second)
//                       X2=ending             X3=pooled_ln
// ---------------------------------------------------------------------------
__global__ __launch_bounds__(256) void finalize_x_kernel(
    const float* __restrict__ part, const int* __restrict__ Lin,
    const float* __restrict__ ln_g, const float* __restrict__ ln_b,
    float* __restrict__ X) {
    __shared__ float r1[256], r2[256];
    const int b = blockIdx.x, tid = threadIdx.x;

    const float4* p4 = (const float4*)part;
    float4 sums[4];
#pragma unroll
    for (int w = 0; w < 4; ++w) {
        float4 s = {0, 0, 0, 0};
        for (int seg = 0; seg < NSEG; ++seg) {
            const float4 v = p4[(size_t)((seg * B_ + b) * 4 + w) * (H_ / 4) + tid];
            s.x += v.x; s.y += v.y; s.z += v.z; s.w += v.w;
        }
        sums[w] = s;
    }

    const int L   = Lin[b];
    const int mid = L >> 1;
    const int Lm1 = L - 1;
    int st = L - 64; if (st < 1) st = 1;
    int cF = mid - 1;  if (cF < 0) cF = 0;
    int cS = Lm1 - mid; if (cS < 0) cS = 0;
    int cE = Lm1 - st;  if (cE < 0) cE = 0;
    const float invP = 1.0f / fmaxf((float)L, 1e-9f);
    const float invF = 1.0f / fmaxf((float)cF, 1e-9f);
    const float invS = 1.0f / fmaxf((float)cS, 1e-9f);
    const float invE = 1.0f / fmaxf((float)cE, 1e-9f);

    float4 pooled;
    pooled.x = sums[0].x * invP; pooled.y = sums[0].y * invP;
    pooled.z = sums[0].z * invP; pooled.w = sums[0].w * invP;

    // LayerNorm statistics over H
    r1[tid] = pooled.x + pooled.y + pooled.z + pooled.w;
    r2[tid] = pooled.x * pooled.x + pooled.y * pooled.y +
              pooled.z * pooled.z + pooled.w * pooled.w;
    __syncthreads();
    for (int off = 128; off > 0; off >>= 1) {
        if (tid < off) { r1[tid] += r1[tid + off]; r2[tid] += r2[tid + off]; }
        __syncthreads();
    }
    const float mu   = r1[0] * (1.0f / (float)H_);
    const float var  = r2[0] * (1.0f / (float)H_) - mu * mu;
    const float rstd = rsqrtf(var + 1e-5f);

    const int h = tid * 4;
    const float4 g  = *(const float4*)(ln_g + h);
    const float4 be = *(const float4*)(ln_b + h);
    float4 pln;
    pln.x = g.x * (pooled.x - mu) * rstd + be.x;
    pln.y = g.y * (pooled.y - mu) * rstd + be.y;
    pln.z = g.z * (pooled.z - mu) * rstd + be.z;
    pln.w = g.w * (pooled.w - mu) * rstd + be.w;

    float4 first, second, ending;
    first.x  = sums[1].x * invF; first.y  = sums[1].y * invF;
    first.z  = sums[1].z * invF; first.w  = sums[1].w * invF;
    second.x = sums[2].x * invS; second.y = sums[2].y * invS;
    second.z = sums[2].z * invS; second.w = sums[2].w * invS;
    ending.x = sums[3].x * invE; ending.y = sums[3].y * invE;
    ending.z = sums[3].z * invE; ending.w = sums[3].w * invE;

    float4 x0, x1;
    x0.x = fmaxf(second.x - first.x, 0.0f); x0.y = fmaxf(second.y - first.y, 0.0f);
    x0.z = fmaxf(second.z - first.z, 0.0f); x0.w = fmaxf(second.w - first.w, 0.0f);
    x1.x = fmaxf(first.x - second.x, 0.0f); x1.y = fmaxf(first.y - second.y, 0.0f);
    x1.z = fmaxf(first.z - second.z, 0.0f); x1.w = fmaxf(first.w - second.w, 0.0f);

    float4* X4 = (float4*)X;
    X4[(size_t)(0 * B_ + b) * (H_ / 4) + tid] = x0;
    X4[(size_t)(1 * B_ + b) * (H_ / 4) + tid] = x1;
    X4[(size_t)(2 * B_ + b) * (H_ / 4) + tid] = ending;
    X4[(size_t)(3 * B_ + b) * (H_ / 4) + tid] = pln;
}

// ---------------------------------------------------------------------------
// K3: head GEMM via V_WMMA_F32_16X16X4_F32.
//   D(16x16 tile of 16x128) = X(16x1024) * w1(1024x128), one wave per N-tile.
//   Epilogue: +b1, exact GELU, *w2, reduce N within tile -> rows[m][ntile].
// VGPR layouts per CDNA5 ISA 7.12.2:
//   A 16x4 f32 : lanes 0-15 M=lane {K=k0,k0+1}; lanes 16-31 M=lane-16 {K=k0+2,k0+3}
//   C 16x16 f32: VGPR r -> (M=r, N=lane) lanes 0-15 ; (M=r+8, N=lane-16) lanes 16-31
// ---------------------------------------------------------------------------
__global__ __launch_bounds__(32) void head_gemm_wmma(
    const float* __restrict__ X,    // 16 x 1024 row-major (this head's input)
    const float* __restrict__ w1,   // 1024 x 128 row-major
    const float* __restrict__ b1,   // 128
    const float* __restrict__ w2,   // 128
    float* __restrict__ rows) {     // 16 x 8 partial row sums (this head)
    const int ntile = blockIdx.x;       // 0..7 -> N block of 16
    const int lane  = threadIdx.x;      // 0..31
    const int m     = lane & 15;
    const int kh    = lane >> 4;        // half selector for K
    const int n     = ntile * 16 + (lane & 15);

    v8f acc = {0.f, 0.f, 0.f, 0.f, 0.f, 0.f, 0.f, 0.f};
    const float* xrow = X + m * H_ + 2 * kh;        // K index = k0 + 2*kh
    const float* wcol = w1 + n + (2 * kh) * 128;
    for (int k0 = 0; k0 < H_; k0 += 4) {
        v2f a, bb;
        a.x  = xrow[k0];
        a.y  = xrow[k0 + 1];
        bb.x = wcol[k0 * 128];
        bb.y = wcol[k0 * 128 + 128];
        acc = __builtin_amdgcn_wmma_f32_16x16x4_f32(
            /*neg_a=*/false, a, /*neg_b=*/false, bb,
            /*c_mod=*/(short)0, acc, /*reuse_a=*/false, /*reuse_b=*/false);
    }

    const float b1n = b1[n];
    const float w2n = w2[n];
    float p[8];
#pragma unroll
    for (int r = 0; r < 8; ++r) {
        float v = acc[r] + b1n;
        p[r] = gelu_exact(v) * w2n;
    }
    // reduce over the 16 N-lanes of each half-wave (rows differ between halves)
#pragma unroll
    for (int r = 0; r < 8; ++r) {
#pragma unroll
        for (int off = 1; off < 16; off <<= 1) p[r] += __shfl_xor(p[r], off, 16);
    }
    if ((lane & 15) == 0) {
#pragma unroll
        for (int r = 0; r < 8; ++r) {
            const int mr = r + 8 * kh;  // lane 0 -> M 0..7, lane 16 -> M 8..15
            rows[mr * 8 + ntile] = p[r];
        }
    }
}

// ---------------------------------------------------------------------------
// K4: assemble combined (1028) and run the fc head. One block per batch.
// ---------------------------------------------------------------------------
__global__ __launch_bounds__(256) void final_kernel(
    const float* __restrict__ X,      // [4][16][1024]; X[3] = pooled_ln
    const float* __restrict__ rows,   // [4][16][8]
    const float* __restrict__ b2e, const float* __restrict__ b2r,
    const float* __restrict__ b2n, const float* __restrict__ b2t,
    const float* __restrict__ fcw1, const float* __restrict__ fcb1,
    const float* __restrict__ fcw2, const float* __restrict__ fcb2,
    float* __restrict__ out) {
    __shared__ float comb[H_ + 4];
    __shared__ float tj[256];
    const int b = blockIdx.x, tid = threadIdx.x;

    for (int i = tid; i < H_; i += 256)
        comb[i] = X[(size_t)(3 * B_ + b) * H_ + i];
    if (tid < 4) {
        float s = (tid == 0) ? b2e[0] : (tid == 1) ? b2r[0]
                 : (tid == 2) ? b2n[0] : b2t[0];
#pragma unroll
        for (int nt = 0; nt < 8; ++nt) s += rows[(tid * B_ + b) * 8 + nt];
        comb[H_ + tid] = s;   // esc, res, end, thr
    }
    __syncthreads();

    float acc = fcb1[tid];
    for (int i = 0; i < H_ + 4; ++i)
        acc = fmaf(comb[i], fcw1[i * 256 + tid], acc);
    tj[tid] = gelu_exact(acc);
    __syncthreads();

    if (tid < 5) {
        float o = fcb2[tid];
        for (int j = 0; j < 256; ++j) o = fmaf(tj[j], fcw2[j * 5 + tid], o);
        out[b * 5 + tid] = o;
    }
}

// ---------------------------------------------------------------------------
// launch
// ---------------------------------------------------------------------------
extern "C" void kernel_launch(void* const* d_in, const int* in_sizes, int n_in,
                              void* d_out, int out_size, void* d_ws, size_t ws_size,
                              hipStream_t stream) {
    const float* hidden = (const float*)d_in[0];
    const int*   mask   = (const int*)d_in[1];
    const float* ln_g   = (const float*)d_in[2];
    const float* ln_b   = (const float*)d_in[3];
    // heads in order esc,res,end,thr; each contributes w1,b1,w2,b2
    const float* w1s[4] = {(const float*)d_in[4],  (const float*)d_in[8],
                           (const float*)d_in[12], (const float*)d_in[16]};
    const float* b1s[4] = {(const float*)d_in[5],  (const float*)d_in[9],
                           (const float*)d_in[13], (const float*)d_in[17]};
    const float* w2s[4] = {(const float*)d_in[6],  (const float*)d_in[10],
                           (const float*)d_in[14], (const float*)d_in[18]};
    const float* b2s[4] = {(const float*)d_in[7],  (const float*)d_in[11],
                           (const float*)d_in[15], (const float*)d_in[19]};
    const float* fcw1 = (const float*)d_in[20];
    const float* fcb1 = (const float*)d_in[21];
    const float* fcw2 = (const float*)d_in[22];
    const float* fcb2 = (const float*)d_in[23];

    float* ws   = (float*)d_ws;
    float* part = ws;                                   // NSEG*B*4*H   = 1048576 f
    float* X    = part + (size_t)NSEG * B_ * 4 * H_;    // 4*B*H        = 65536 f
    float* rows = X + 4 * B_ * H_;                      // 4*16*8       = 512 f
    int*   Lint = (int*)(rows + 4 * B_ * 8);            // 16 ints

    lengths_kernel<<<B_, 256, 0, stream>>>(mask, Lint);
    partial_sums_kernel<<<dim3(NSEG, B_), 256, 0, stream>>>(hidden, mask, Lint, part);
    finalize_x_kernel<<<B_, 256, 0, stream>>>(part, Lint, ln_g, ln_b, X);
    for (int hd = 0; hd < 4; ++hd)
        head_gemm_wmma<<<8, 32, 0, stream>>>(X + (size_t)hd * B_ * H_, w1s[hd],
                                             b1s[hd], w2s[hd], rows + hd * B_ * 8);
    final_kernel<<<B_, 256, 0, stream>>>(X, rows, b2s[0], b2s[1], b2s[2], b2s[3],
                                         fcw1, fcb1, fcw2, fcb2, (float*)d_out);
}